// PillarPFNLayer_44092134261307
// MI455X (gfx1250) — compile-verified
//
#include <hip/hip_runtime.h>
#include <hip/hip_bf16.h>

#define N_POINTS    2000000
#define IN_CH       10
#define WP_STRIDE   12            // K padded to 12 (3 chunks of 4)
#define OUT_CH      32
#define NUM_PILLARS 100000
#define BN_EPS      1e-3f

#define NSTAT        65           // 10 sums + 55 upper-tri second moments
#define STAT_BLOCKS  1024
#define STAT_THREADS 256

#define TILE_BLOCKS  15625        // 15625 blocks * 8 waves * 16 rows = 2,000,000

typedef __attribute__((ext_vector_type(2))) float v2f;
typedef __attribute__((ext_vector_type(8))) float v8f;

// ---------------------------------------------------------------------------
// Pass 1: per-block partial sums of x (10) and upper-tri x⊗x (55)
// ---------------------------------------------------------------------------
__global__ void stats_partial(const float* __restrict__ in,
                              float* __restrict__ partials) {
  float acc[NSTAT];
#pragma unroll
  for (int q = 0; q < NSTAT; ++q) acc[q] = 0.0f;

  const int tid    = blockIdx.x * blockDim.x + threadIdx.x;
  const int stride = gridDim.x * blockDim.x;
  for (int r = tid; r < N_POINTS; r += stride) {
    float x[IN_CH];
    const float* row = in + (size_t)r * IN_CH;
#pragma unroll
    for (int i = 0; i < IN_CH; ++i) x[i] = row[i];
#pragma unroll
    for (int i = 0; i < IN_CH; ++i) acc[i] += x[i];
    int q = IN_CH;
#pragma unroll
    for (int i = 0; i < IN_CH; ++i)
#pragma unroll
      for (int j = i; j < IN_CH; ++j) acc[q++] += x[i] * x[j];
  }

  // wave32 shuffle reduction
#pragma unroll
  for (int off = 16; off > 0; off >>= 1)
#pragma unroll
    for (int q = 0; q < NSTAT; ++q) acc[q] += __shfl_down(acc[q], off);

  __shared__ float red[STAT_THREADS / 32][NSTAT];
  const int wid  = threadIdx.x >> 5;
  const int lane = threadIdx.x & 31;
  if (lane == 0) {
#pragma unroll
    for (int q = 0; q < NSTAT; ++q) red[wid][q] = acc[q];
  }
  __syncthreads();
  if (threadIdx.x < NSTAT) {
    float t = 0.0f;
#pragma unroll
    for (int w = 0; w < STAT_THREADS / 32; ++w) t += red[w][threadIdx.x];
    partials[(size_t)blockIdx.x * NSTAT + threadIdx.x] = t;
  }
}

// ---------------------------------------------------------------------------
// Pass 1b: reduce partials; fold BatchNorm into the linear weights
//   mean_c = w_c . E[x] + b_c ;  var_c = w_c^T Cov(x) w_c
//   Wp = W * (gamma * rsqrt(var+eps)) ;  bp = (b - mean)*sg + beta
// ---------------------------------------------------------------------------
__global__ void finalize(const float* __restrict__ partials,
                         const float* __restrict__ W,
                         const float* __restrict__ b,
                         const float* __restrict__ gamma,
                         const float* __restrict__ beta,
                         float* __restrict__ Wp,
                         float* __restrict__ bp) {
  __shared__ float stat[NSTAT];
  const int t = threadIdx.x;
  if (t < NSTAT) {
    float a = 0.0f;
    for (int bk = 0; bk < STAT_BLOCKS; ++bk) a += partials[(size_t)bk * NSTAT + t];
    stat[t] = a;
  }
  __syncthreads();
  if (t < OUT_CH) {
    const float invN = 1.0f / (float)N_POINTS;
    float mx[IN_CH], w[IN_CH];
#pragma unroll
    for (int i = 0; i < IN_CH; ++i) mx[i] = stat[i] * invN;
#pragma unroll
    for (int i = 0; i < IN_CH; ++i) w[i] = W[t * IN_CH + i];

    float mean = b[t];
#pragma unroll
    for (int i = 0; i < IN_CH; ++i) mean += w[i] * mx[i];

    float var = 0.0f;
    int q = IN_CH;
#pragma unroll
    for (int i = 0; i < IN_CH; ++i)
#pragma unroll
      for (int j = i; j < IN_CH; ++j) {
        float cov = stat[q] * invN - mx[i] * mx[j];
        var += ((i == j) ? 1.0f : 2.0f) * w[i] * w[j] * cov;
        ++q;
      }

    const float sg = gamma[t] * rsqrtf(var + BN_EPS);
#pragma unroll
    for (int i = 0; i < WP_STRIDE; ++i)
      Wp[t * WP_STRIDE + i] = (i < IN_CH) ? w[i] * sg : 0.0f;
    bp[t] = (b[t] - mean) * sg + beta[t];
  }
}

// ---------------------------------------------------------------------------
// Zero the pillar-max table (+0.0 == uint 0; valid floor since ReLU >= 0)
// ---------------------------------------------------------------------------
__global__ void init_pmax(unsigned* __restrict__ pmax) {
  const int i = blockIdx.x * blockDim.x + threadIdx.x;
  if (i < NUM_PILLARS * OUT_CH) pmax[i] = 0u;
}

// ---------------------------------------------------------------------------
// One 16-row tile of relu((x @ Wp^T) + bp) via V_WMMA_F32_16X16X4_F32.
// A (16x4 f32): lanes 0-15 hold K=k0,k0+1 ; lanes 16-31 hold K=k0+2,k0+3
// B (4x16 f32): mirrored, lane index = N
// C/D (16x16 f32, 8 VGPRs): vgpr j, lanes 0-15 -> M=j ; lanes 16-31 -> M=j+8
// ---------------------------------------------------------------------------
__device__ __forceinline__ void compute_tile(const float* __restrict__ in,
                                             const float* __restrict__ Wp,
                                             const float* __restrict__ bp,
                                             int row0, int lane,
                                             v8f& c0, v8f& c1) {
  const int half = lane >> 4;   // 0 or 1
  const int l    = lane & 15;

  c0 = (v8f){0.f, 0.f, 0.f, 0.f, 0.f, 0.f, 0.f, 0.f};
  c1 = (v8f){0.f, 0.f, 0.f, 0.f, 0.f, 0.f, 0.f, 0.f};

#pragma unroll
  for (int kk = 0; kk < 3; ++kk) {
    const int kb = kk * 4 + half * 2;

    v2f a;
    if (kb < IN_CH) {                       // kb==10 (chunk 2, upper half) -> pad 0
      const float* rp = in + (size_t)(row0 + l) * IN_CH + kb;
      a = (v2f){rp[0], rp[1]};
    } else {
      a = (v2f){0.f, 0.f};
    }

    // B[k][n] = Wp[n][k]; Wp is padded to 12 cols, so kb up to 10/11 reads zeros
    const float* w0 = Wp + (size_t)l * WP_STRIDE + kb;
    const float* w1 = Wp + (size_t)(16 + l) * WP_STRIDE + kb;
    v2f b0 = (v2f){w0[0], w0[1]};
    v2f b1 = (v2f){w1[0], w1[1]};

    c0 = __builtin_amdgcn_wmma_f32_16x16x4_f32(false, a, false, b0, (short)0, c0,
                                               false, false);
    c1 = __builtin_amdgcn_wmma_f32_16x16x4_f32(false, a, false, b1, (short)0, c1,
                                               false, false);
  }

  const float bb0 = bp[l];
  const float bb1 = bp[16 + l];
#pragma unroll
  for (int j = 0; j < 8; ++j) {
    c0[j] = fmaxf(c0[j] + bb0, 0.0f);
    c1[j] = fmaxf(c1[j] + bb1, 0.0f);
  }
}

// ---------------------------------------------------------------------------
// Pass 2: compute x, scatter atomicMax into per-pillar maxima
// ---------------------------------------------------------------------------
__global__ void pass2(const float* __restrict__ in,
                      const int* __restrict__ unq,
                      const float* __restrict__ Wp,
                      const float* __restrict__ bp,
                      unsigned* __restrict__ pmax) {
  const int gtid = blockIdx.x * blockDim.x + threadIdx.x;
  const int wave = gtid >> 5;
  const int lane = threadIdx.x & 31;
  const int row0 = wave * 16;

  v8f c0, c1;
  compute_tile(in, Wp, bp, row0, lane, c0, c1);

  const int half = lane >> 4;
  const int l    = lane & 15;
#pragma unroll
  for (int j = 0; j < 8; ++j) {
    const int r = row0 + j + half * 8;
    const int p = unq[r];
    unsigned* base = pmax + (size_t)p * OUT_CH;
    atomicMax(base + l,      __float_as_uint(c0[j]));
    atomicMax(base + 16 + l, __float_as_uint(c1[j]));
  }
}

// ---------------------------------------------------------------------------
// Pass 3: recompute x (input stays L2-resident; cheaper than spilling 256MB),
// gather pillar max, average, and stream the 256MB result with NT stores so
// the output never evicts the input / pmax working set from L2.
// ---------------------------------------------------------------------------
__global__ void pass3(const float* __restrict__ in,
                      const int* __restrict__ unq,
                      const float* __restrict__ Wp,
                      const float* __restrict__ bp,
                      const unsigned* __restrict__ pmax,
                      float* __restrict__ out) {
  const int gtid = blockIdx.x * blockDim.x + threadIdx.x;
  const int wave = gtid >> 5;
  const int lane = threadIdx.x & 31;
  const int row0 = wave * 16;

  v8f c0, c1;
  compute_tile(in, Wp, bp, row0, lane, c0, c1);

  const int half = lane >> 4;
  const int l    = lane & 15;
#pragma unroll
  for (int j = 0; j < 8; ++j) {
    const int r = row0 + j + half * 8;
    const int p = unq[r];
    const unsigned* base = pmax + (size_t)p * OUT_CH;
    const float m0 = __uint_as_float(base[l]);
    const float m1 = __uint_as_float(base[16 + l]);
    __builtin_nontemporal_store((c0[j] + m0) * 0.5f, out + (size_t)r * OUT_CH + l);
    __builtin_nontemporal_store((c1[j] + m1) * 0.5f, out + (size_t)r * OUT_CH + 16 + l);
  }
}

// ---------------------------------------------------------------------------
extern "C" void kernel_launch(void* const* d_in, const int* in_sizes, int n_in,
                              void* d_out, int out_size, void* d_ws, size_t ws_size,
                              hipStream_t stream) {
  (void)in_sizes; (void)n_in; (void)out_size; (void)ws_size;

  const float* in    = (const float*)d_in[0];
  const float* W     = (const float*)d_in[1];
  const float* b     = (const float*)d_in[2];
  const float* gamma = (const float*)d_in[3];
  const float* beta  = (const float*)d_in[4];
  const int*   unq   = (const int*)d_in[5];
  float*       out   = (float*)d_out;

  // workspace layout (floats): partials | Wp(32x12) | bp(32) | pmax(100000x32)
  float* partials = (float*)d_ws;
  float* Wp       = partials + (size_t)STAT_BLOCKS * NSTAT;
  float* bp       = Wp + OUT_CH * WP_STRIDE;
  unsigned* pmax  = (unsigned*)(bp + OUT_CH);

  stats_partial<<<STAT_BLOCKS, STAT_THREADS, 0, stream>>>(in, partials);
  finalize<<<1, 256, 0, stream>>>(partials, W, b, gamma, beta, Wp, bp);
  init_pmax<<<(NUM_PILLARS * OUT_CH + 255) / 256, 256, 0, stream>>>(pmax);
  pass2<<<TILE_BLOCKS, 256, 0, stream>>>(in, unq, Wp, bp, pmax);
  pass3<<<TILE_BLOCKS, 256, 0, stream>>>(in, unq, Wp, bp, pmax, out);
}